// SymbioseGNN_82377472737678
// MI455X (gfx1250) — compile-verified
//
#include <hip/hip_runtime.h>

typedef float v2f __attribute__((ext_vector_type(2)));
typedef float v8f __attribute__((ext_vector_type(8)));

#define N_NODES 50000
#define FDIM    128
#define N_EDGES 800000
#define KSTRIDE 260   // transposed-weight LDS k-stride: even (8B align) and ≡4 mod 64 (conflict-free)

// ---------------- zero-fill ----------------
__global__ __launch_bounds__(256) void sage_zero_kernel(float* __restrict__ p, int n) {
    int i = blockIdx.x * 256 + threadIdx.x;
    if (i < n) p[i] = 0.0f;
}

// ---------------- edge scatter-add (mean numerator + degree) ----------------
// one wave (32 lanes) per edge; each lane moves 16B (float4) and issues 4 f32 atomics
__global__ __launch_bounds__(256) void sage_scatter_kernel(
    const float* __restrict__ x,
    const int*   __restrict__ src,
    const int*   __restrict__ dst,
    float* __restrict__ msg,
    float* __restrict__ cnt,
    int with_count)
{
    int lane = threadIdx.x & 31;
    int wave = threadIdx.x >> 5;
    int e = blockIdx.x * 8 + wave;
    if (e >= N_EDGES) return;
    int s = src[e];
    int d = dst[e];
    const float4* xs = (const float4*)(x + (long long)s * FDIM);
    float4 v = xs[lane];                       // 32 lanes * 16B = full 128-f row, coalesced
    float* o = msg + (long long)d * FDIM + lane * 4;
    atomicAdd(o + 0, v.x);                     // no-return f32 atomics, L2-resident target
    atomicAdd(o + 1, v.y);
    atomicAdd(o + 2, v.z);
    atomicAdd(o + 3, v.w);
    if (with_count && lane == 0) atomicAdd(cnt + d, 1.0f);
}

// ---------------- fused mean-normalize + dual GEMM + bias (+relu) ----------------
// out[i,:] = act( (msg[i,:]/max(cnt,1)) @ Wl + x[i,:] @ Wr + b )
// block = 8 waves; each wave: 16x128 output tile via V_WMMA_F32_16X16X4_F32, K=256
// weights staged TRANSPOSED in LDS so every B fragment is one conflict-free ds_load_b64
__global__ __launch_bounds__(256) void sage_gemm_kernel(
    const float* __restrict__ msg, const float* __restrict__ cnt,
    const float* __restrict__ x,
    const float* __restrict__ Wl,  const float* __restrict__ Wr,
    const float* __restrict__ bias,
    float* __restrict__ out, int do_relu)
{
    __shared__ float lw[128 * KSTRIDE];        // lw[n*KSTRIDE + k] = Wfused[k][n], 133 KB

    int tid = threadIdx.x;
    for (int idx = tid; idx < 256 * 128; idx += 256) {
        int k = idx >> 7;                      // fused-K index (0..255)
        int n = idx & 127;                     // output column
        float w = (k < 128) ? Wl[idx] : Wr[idx - 128 * 128];
        lw[n * KSTRIDE + k] = w;               // transposed store (one-time, coalesced reads)
    }
    __syncthreads();

    int lane = tid & 31;
    int wave = tid >> 5;
    int h    = lane >> 4;                      // half-wave select: k offset +2 (ISA A/B layout)
    int col  = lane & 15;

    int rowBase = blockIdx.x * 128 + wave * 16;
    int rowA    = rowBase + col;               // this lane's fixed A row
    int rowAc   = rowA < N_NODES ? rowA : (N_NODES - 1);

    float scale = 1.0f / fmaxf(cnt[rowAc], 1.0f);
    const float* arow_m = msg + (long long)rowAc * FDIM + 2 * h;
    const float* arow_x = x   + (long long)rowAc * FDIM + 2 * h;
    const float* brow   = &lw[col * KSTRIDE + 2 * h];  // + t*16*KSTRIDE + kk per fragment

    v8f acc[8];
    v8f z = {0.f, 0.f, 0.f, 0.f, 0.f, 0.f, 0.f, 0.f};
#pragma unroll
    for (int t = 0; t < 8; ++t) acc[t] = z;

    // ---- K half 1: mean features (k = 0..127), branch-free ----
#pragma unroll 4
    for (int kk = 0; kk < 128; kk += 4) {
        v2f a = *(const v2f*)(arow_m + kk);
        a.x *= scale;
        a.y *= scale;
#pragma unroll
        for (int t = 0; t < 8; ++t) {
            v2f b = *(const v2f*)(brow + t * (16 * KSTRIDE) + kk);   // ds_load_b64
            acc[t] = __builtin_amdgcn_wmma_f32_16x16x4_f32(
                false, a, false, b, (short)0, acc[t], false, false);
        }
    }

    // ---- K half 2: self features (k = 128..255), branch-free ----
#pragma unroll 4
    for (int kk = 0; kk < 128; kk += 4) {
        v2f a = *(const v2f*)(arow_x + kk);
#pragma unroll
        for (int t = 0; t < 8; ++t) {
            v2f b = *(const v2f*)(brow + t * (16 * KSTRIDE) + 128 + kk);
            acc[t] = __builtin_amdgcn_wmma_f32_16x16x4_f32(
                false, a, false, b, (short)0, acc[t], false, false);
        }
    }

    // D layout: lane 0-15 -> M=v, N=lane ; lane 16-31 -> M=v+8, N=lane-16
#pragma unroll
    for (int t = 0; t < 8; ++t) {
        int n  = t * 16 + col;
        float bv = bias[n];
#pragma unroll
        for (int v = 0; v < 8; ++v) {
            int row = rowBase + v + 8 * h;
            float val = acc[t][v] + bv;
            if (do_relu) val = fmaxf(val, 0.0f);
            if (row < N_NODES) out[(long long)row * FDIM + n] = val;
        }
    }
}

extern "C" void kernel_launch(void* const* d_in, const int* in_sizes, int n_in,
                              void* d_out, int out_size, void* d_ws, size_t ws_size,
                              hipStream_t stream) {
    const float* x   = (const float*)d_in[0];
    const int*   ei  = (const int*)d_in[1];
    const float* Wl0 = (const float*)d_in[2];
    const float* Wr0 = (const float*)d_in[3];
    const float* b0  = (const float*)d_in[4];
    const float* Wl1 = (const float*)d_in[5];
    const float* Wr1 = (const float*)d_in[6];
    const float* b1  = (const float*)d_in[7];
    float* out = (float*)d_out;

    const int* src = ei;
    const int* dst = ei + N_EDGES;

    // workspace: msg[50000*128] | cnt[50000] | h[50000*128]  (~51.4 MB)
    float* msg = (float*)d_ws;
    float* cnt = msg + (size_t)N_NODES * FDIM;
    float* hbuf = cnt + N_NODES;

    const int nmsg = N_NODES * FDIM;
    const int gemm_blocks = (N_NODES + 127) / 128;   // 391
    const int edge_blocks = N_EDGES / 8;             // 100000

    // ---- layer 0 ----
    sage_zero_kernel<<<(nmsg + N_NODES + 255) / 256, 256, 0, stream>>>(msg, nmsg + N_NODES);
    sage_scatter_kernel<<<edge_blocks, 256, 0, stream>>>(x, src, dst, msg, cnt, 1);
    sage_gemm_kernel<<<gemm_blocks, 256, 0, stream>>>(msg, cnt, x, Wl0, Wr0, b0, hbuf, 1);

    // ---- layer 1 (cnt is identical: same dst vector) ----
    sage_zero_kernel<<<(nmsg + 255) / 256, 256, 0, stream>>>(msg, nmsg);
    sage_scatter_kernel<<<edge_blocks, 256, 0, stream>>>(hbuf, src, dst, msg, cnt, 0);
    sage_gemm_kernel<<<gemm_blocks, 256, 0, stream>>>(msg, cnt, hbuf, Wl1, Wr1, b1, out, 0);
}